// AttentionRegression_80771154969106
// MI455X (gfx1250) — compile-verified
//
#include <hip/hip_runtime.h>
#include <math.h>

typedef __attribute__((ext_vector_type(2))) float v2f;
typedef __attribute__((ext_vector_type(8))) float v8f;

#define NFEAT 128
#define HDIM  12
#define TILE_M 16
#define WAVES_PER_BLOCK 8
#define KCHUNKS (NFEAT / 4)   // 32 chained V_WMMA_F32_16X16X4_F32 per tile

// ---------------------------------------------------------------------------
// Pass 1: stream neighbours once; per 16-row tile compute
//   C[16x16] = rows[16x128] @ Bmat[128x16] via 32x v_wmma_f32_16x16x4_f32
// where Bmat cols 0..11 = f_W1[:,1:].T, col 12 = g_W[1:], cols 13..15 = 0.
// Epilogue: add f_x column term + bias, tanh, W2-weighted lane reduction,
// sigmoid -> s[t]; col 12 of C -> q[t] = neighbours[t] . g_W[1:].
// ---------------------------------------------------------------------------
__global__ __launch_bounds__(256) void attn_pass1(
    const float* __restrict__ neighbours,
    const int*   __restrict__ seg_ids,
    const float* __restrict__ f_x,
    const float* __restrict__ f_W1,
    const float* __restrict__ f_b1,
    const float* __restrict__ f_W2,
    const float* __restrict__ f_b2,
    const float* __restrict__ g_W,
    float* __restrict__ s_buf,
    float* __restrict__ q_buf,
    int n_tiles)
{
    __shared__ v2f Bfrag[KCHUNKS * 32];

    // Build B fragments once per block (same for every tile).
    for (int e = threadIdx.x; e < KCHUNKS * 32; e += blockDim.x) {
        int c  = e >> 5;
        int l  = e & 31;
        int kh = (l >> 4) & 1;
        int nn = l & 15;
        int k0 = 4 * c + 2 * kh;          // global K index of reg0
        float b0, b1;
        if (nn < HDIM) {
            b0 = f_W1[nn * (NFEAT + 1) + 1 + k0];
            b1 = f_W1[nn * (NFEAT + 1) + 2 + k0];
        } else if (nn == HDIM) {
            b0 = g_W[1 + k0];
            b1 = g_W[2 + k0];
        } else {
            b0 = 0.0f; b1 = 0.0f;
        }
        v2f v; v.x = b0; v.y = b1;
        Bfrag[e] = v;
    }
    __syncthreads();

    const int wave = threadIdx.x >> 5;
    const int lane = threadIdx.x & 31;
    const int tile = blockIdx.x * WAVES_PER_BLOCK + wave;
    if (tile >= n_tiles) return;          // wave-uniform: EXEC stays all-1s

    const int  n     = lane & 15;         // output column / row-for-loading
    const int  khalf = lane >> 4;
    const long rbase = (long)tile * TILE_M;
    const float* rowp = neighbours + (rbase + n) * NFEAT + 2 * khalf;

    v8f acc = {0.f, 0.f, 0.f, 0.f, 0.f, 0.f, 0.f, 0.f};
    #pragma unroll 4
    for (int c = 0; c < KCHUNKS; ++c) {
        v2f a  = __builtin_nontemporal_load((const v2f*)(rowp + 4 * c));
        v2f bv = Bfrag[c * 32 + lane];
        acc = __builtin_amdgcn_wmma_f32_16x16x4_f32(
            /*neg_a=*/false, a, /*neg_b=*/false, bv,
            /*c_mod=*/(short)0, acc, /*reuse_a=*/false, /*reuse_b=*/false);
    }

    // Epilogue. Lane 'n' (<16) fetched row n's fx; broadcast per output row.
    const int   sid  = seg_ids[rbase + n];
    const float fxv  = f_x[sid];
    const float w1c0 = (n < HDIM) ? f_W1[n * (NFEAT + 1)] : 0.0f;
    const float b1n  = (n < HDIM) ? f_b1[n] : 0.0f;
    const float w2n  = (n < HDIM) ? f_W2[n] : 0.0f;
    const float fb2  = f_b2[0];

    #pragma unroll
    for (int i = 0; i < 8; ++i) {
        const int m = khalf * 8 + i;                   // row handled this step
        float fxr = __shfl(fxv, m, 32);                // lane m holds row m's fx
        float pre = acc[i] + w1c0 * fxr + b1n;
        float contrib = tanhf(pre) * w2n;              // 0 for lanes n>=12
        // sum over the 16-lane half (masks < 16 keep lanes within the half)
        contrib += __shfl_xor(contrib, 1, 32);
        contrib += __shfl_xor(contrib, 2, 32);
        contrib += __shfl_xor(contrib, 4, 32);
        contrib += __shfl_xor(contrib, 8, 32);
        float sval = 1.0f / (1.0f + expf(-(contrib + fb2)));
        if (n == HDIM) {                               // lane owning column 12
            long t = rbase + m;
            s_buf[t] = sval;
            q_buf[t] = acc[i];                         // neighbours[t].g_W[1:]
        }
    }
}

// ---------------------------------------------------------------------------
// Pass 1b: segment boundaries from sorted seg_ids.
// seg_start[b] = first t with seg_ids[t] >= b;  seg_start[B] = T.
// ---------------------------------------------------------------------------
__global__ __launch_bounds__(256) void seg_bounds(
    const int* __restrict__ seg_ids, int* __restrict__ seg_start, int T, int B)
{
    int t = blockIdx.x * blockDim.x + threadIdx.x;
    if (t >= T) return;
    int c = seg_ids[t];
    int p = (t == 0) ? -1 : seg_ids[t - 1];
    for (int b = p + 1; b <= c; ++b) seg_start[b] = t;
    if (t == T - 1) {
        for (int b = c + 1; b <= B; ++b) seg_start[b] = T;
    }
}

// ---------------------------------------------------------------------------
// Pass 2: one wave per segment: softmax-combine of (s, q) streams.
// out[b] = g_W[0]*f_x[b] + g_b + sum_t softmax(s)_t * q_t
// ---------------------------------------------------------------------------
__global__ __launch_bounds__(256) void attn_pass2(
    const float* __restrict__ s_buf,
    const float* __restrict__ q_buf,
    const int*   __restrict__ seg_start,
    const float* __restrict__ f_x,
    const float* __restrict__ g_W,
    const float* __restrict__ g_b,
    float* __restrict__ out,
    int B)
{
    int b = blockIdx.x * (blockDim.x >> 5) + (threadIdx.x >> 5);
    if (b >= B) return;
    int lane = threadIdx.x & 31;
    int st = seg_start[b];
    int en = seg_start[b + 1];

    float m = -3.402823466e+38f;
    for (int t = st + lane; t < en; t += 32) m = fmaxf(m, s_buf[t]);
    m = fmaxf(m, __shfl_xor(m, 1, 32));
    m = fmaxf(m, __shfl_xor(m, 2, 32));
    m = fmaxf(m, __shfl_xor(m, 4, 32));
    m = fmaxf(m, __shfl_xor(m, 8, 32));
    m = fmaxf(m, __shfl_xor(m, 16, 32));

    float zsum = 0.0f, zq = 0.0f;
    for (int t = st + lane; t < en; t += 32) {
        float z = expf(s_buf[t] - m);
        zsum += z;
        zq   += z * q_buf[t];
    }
    zsum += __shfl_xor(zsum, 1, 32);  zq += __shfl_xor(zq, 1, 32);
    zsum += __shfl_xor(zsum, 2, 32);  zq += __shfl_xor(zq, 2, 32);
    zsum += __shfl_xor(zsum, 4, 32);  zq += __shfl_xor(zq, 4, 32);
    zsum += __shfl_xor(zsum, 8, 32);  zq += __shfl_xor(zq, 8, 32);
    zsum += __shfl_xor(zsum, 16, 32); zq += __shfl_xor(zq, 16, 32);

    if (lane == 0) {
        float emb = (en > st) ? (zq / zsum) : 0.0f;
        out[b] = g_W[0] * f_x[b] + emb + g_b[0];
    }
}

// ---------------------------------------------------------------------------
extern "C" void kernel_launch(void* const* d_in, const int* in_sizes, int n_in,
                              void* d_out, int out_size, void* d_ws, size_t ws_size,
                              hipStream_t stream) {
    const float* f_x        = (const float*)d_in[0];
    const float* neighbours = (const float*)d_in[1];
    const int*   seg_ids    = (const int*)  d_in[2];
    const float* f_W1       = (const float*)d_in[3];
    const float* f_b1       = (const float*)d_in[4];
    const float* f_W2       = (const float*)d_in[5];
    const float* f_b2       = (const float*)d_in[6];
    const float* g_W        = (const float*)d_in[7];
    const float* g_b        = (const float*)d_in[8];

    const int B = in_sizes[0];        // 16384
    const int T = in_sizes[2];        // 1048576

    float* s_buf     = (float*)d_ws;
    float* q_buf     = s_buf + T;
    int*   seg_start = (int*)(q_buf + T);
    float* out       = (float*)d_out;

    const int n_tiles = T / TILE_M;   // T is a multiple of 16
    dim3 blk(256);

    attn_pass1<<<(n_tiles + WAVES_PER_BLOCK - 1) / WAVES_PER_BLOCK, blk, 0, stream>>>(
        neighbours, seg_ids, f_x, f_W1, f_b1, f_W2, f_b2, g_W, s_buf, q_buf, n_tiles);

    seg_bounds<<<(T + 255) / 256, blk, 0, stream>>>(seg_ids, seg_start, T, B);

    attn_pass2<<<(B + WAVES_PER_BLOCK - 1) / WAVES_PER_BLOCK, blk, 0, stream>>>(
        s_buf, q_buf, seg_start, f_x, g_W, g_b, out, B);
}